// GCLSTM_50654844289081
// MI455X (gfx1250) — compile-verified
//
#include <hip/hip_runtime.h>
#include <hip/hip_bf16.h>

// ---- problem constants (match reference) ----
#define NN      50000
#define EE      800000
#define IN_DIM  128
#define HID     64
#define KDIM    384          // 128 (x) + 128 (Tx) + 64 (h) + 64 (Th)
#define CDIM    256          // 4 gates * 64
#define NEG_SLOPE 0.01f

typedef float v2f __attribute__((ext_vector_type(2)));
typedef float v8f __attribute__((ext_vector_type(8)));

__device__ __forceinline__ float sigm(float x) { return 1.0f / (1.0f + expf(-x)); }

// ------------------------------------------------------------------
// 0) zero a 4-byte-word range (grid-stride)
// ------------------------------------------------------------------
__global__ void k_zero(unsigned* __restrict__ p, long n) {
    long i = (long)blockIdx.x * blockDim.x + threadIdx.x;
    long stride = (long)gridDim.x * blockDim.x;
    for (; i < n; i += stride) p[i] = 0u;
}

// ------------------------------------------------------------------
// 1) pack Wcat (KDIM x CDIM) and bcat (CDIM) from Wx/Wh/bx/bh/bg
//    Wx: (4,2,128,64) Wh: (4,2,64,64) row-major
// ------------------------------------------------------------------
__global__ void k_prep(const float* __restrict__ Wx, const float* __restrict__ Wh,
                       const float* __restrict__ bx, const float* __restrict__ bh,
                       const float* __restrict__ bg,
                       float* __restrict__ Wcat, float* __restrict__ bcat) {
    int t = blockIdx.x * blockDim.x + threadIdx.x;
    if (t >= KDIM * CDIM) return;
    int k = t / CDIM, col = t % CDIM;
    int g = col / HID, j = col % HID;
    float v;
    if (k < 128)       v = Wx[(((g * 2 + 0) * 128) + k)         * 64 + j];
    else if (k < 256)  v = Wx[(((g * 2 + 1) * 128) + (k - 128)) * 64 + j];
    else if (k < 320)  v = Wh[(((g * 2 + 0) * 64)  + (k - 256)) * 64 + j];
    else               v = Wh[(((g * 2 + 1) * 64)  + (k - 320)) * 64 + j];
    Wcat[k * CDIM + col] = v;
    if (k == 0) bcat[col] = bx[g * 64 + j] + bh[g * 64 + j] + bg[g * 64 + j];
}

// ------------------------------------------------------------------
// 2) per-edge: deg[src] += ew (f32 atomic), cnt[dst] += 1 (int atomic)
// ------------------------------------------------------------------
__global__ void k_deg_count(const int* __restrict__ src, const int* __restrict__ dst,
                            const float* __restrict__ ew,
                            float* __restrict__ deg, int* __restrict__ cnt) {
    int e = blockIdx.x * blockDim.x + threadIdx.x;
    if (e >= EE) return;
    atomicAdd(&deg[src[e]], ew[e]);
    atomicAdd(&cnt[dst[e]], 1);
}

// 3) deg -> dinv (in place)
__global__ void k_dinv(float* __restrict__ deg) {
    int n = blockIdx.x * blockDim.x + threadIdx.x;
    if (n >= NN) return;
    float d = deg[n];
    deg[n] = (d > 0.0f) ? rsqrtf(d) : 0.0f;
}

// ------------------------------------------------------------------
// 4) single-block exclusive scan of cnt[0..NN) -> rs[0..NN], copy to cur
// ------------------------------------------------------------------
__global__ void __launch_bounds__(1024)
k_scan(const int* __restrict__ cnt, int* __restrict__ rs, int* __restrict__ cur) {
    __shared__ int sums[1024];
    const int t = threadIdx.x;
    const int chunk = (NN + 1023) / 1024;           // 49
    int beg = t * chunk;
    int end = beg + chunk; if (end > NN) end = NN;
    int s = 0;
    for (int i = beg; i < end; ++i) s += cnt[i];
    sums[t] = s;
    __syncthreads();
    // Hillis-Steele inclusive scan over 1024 partial sums
    for (int off = 1; off < 1024; off <<= 1) {
        int v = (t >= off) ? sums[t - off] : 0;
        __syncthreads();
        sums[t] += v;
        __syncthreads();
    }
    int run = (t == 0) ? 0 : sums[t - 1];           // exclusive prefix for this chunk
    for (int i = beg; i < end; ++i) {
        rs[i] = run; cur[i] = run;
        run += cnt[i];
    }
    if (t == 1023) rs[NN] = run;                    // == EE
}

// ------------------------------------------------------------------
// 5) fill CSR: slot = cur[dst]++; esrc[slot]=src; enrm[slot]=cheb norm
// ------------------------------------------------------------------
__global__ void k_fill(const int* __restrict__ src, const int* __restrict__ dst,
                       const float* __restrict__ ew, const float* __restrict__ dinv,
                       int* __restrict__ cur,
                       int* __restrict__ esrc, float* __restrict__ enrm) {
    int e = blockIdx.x * blockDim.x + threadIdx.x;
    if (e >= EE) return;
    int s = src[e], d = dst[e];
    float w = -dinv[s] * ew[e] * dinv[d];
    int p = atomicAdd(&cur[d], 1);
    esrc[p] = s;
    enrm[p] = w;
}

// ------------------------------------------------------------------
// 6) CSR gather: Tx[n][f] = sum_in-edges w * x[s][f]; Th likewise from h.
//    1 block = 1 node, 192 threads (128 x-features + 64 h-features).
//    Edge (src, w) pairs staged through LDS in chunks of 192.
// ------------------------------------------------------------------
#define GCHUNK 192
__global__ void __launch_bounds__(192)
k_gather(const int* __restrict__ rs, const int* __restrict__ esrc,
         const float* __restrict__ enrm,
         const float* __restrict__ x, const float* __restrict__ h,
         float* __restrict__ Tx, float* __restrict__ Th) {
    __shared__ int   ssrc[GCHUNK];
    __shared__ float swt [GCHUNK];
    const int n = blockIdx.x;
    const int f = threadIdx.x;          // 0..191
    const int beg = rs[n], end = rs[n + 1];
    float acc = 0.0f;
    for (int cb = beg; cb < end; cb += GCHUNK) {
        int m = end - cb; if (m > GCHUNK) m = GCHUNK;
        if (f < m) { ssrc[f] = esrc[cb + f]; swt[f] = enrm[cb + f]; }
        __syncthreads();
        if (f < 128) {
            for (int j = 0; j < m; ++j)
                acc += swt[j] * x[(size_t)ssrc[j] * IN_DIM + f];
        } else {
            const int fh = f - 128;
            for (int j = 0; j < m; ++j)
                acc += swt[j] * h[(size_t)ssrc[j] * HID + fh];
        }
        __syncthreads();
    }
    if (f < 128) Tx[(size_t)n * IN_DIM + f]        = acc;
    else         Th[(size_t)n * HID + (f - 128)]   = acc;
}

// ------------------------------------------------------------------
// 7) fused GEMM (WMMA f32 16x16x4) + LSTM gating + head
//    1 block = 32 nodes (two 16-row M-tiles), 128 threads = 4 waves.
//    Wave w owns gate w's 64 columns; each B fragment feeds 2 WMMAs.
// ------------------------------------------------------------------
#define TILE_M    32
#define AS_STRIDE 388   // 384 + 4 pad -> conflict-free column reads
#define ZS_STRIDE 260   // 256 + 4 pad

__global__ void __launch_bounds__(128)
k_gclstm(const float* __restrict__ x,  const float* __restrict__ Tx,
         const float* __restrict__ h,  const float* __restrict__ Th,
         const float* __restrict__ c,
         const float* __restrict__ Wcat, const float* __restrict__ bcat,
         const float* __restrict__ wc,   const float* __restrict__ Whead,
         const float* __restrict__ bhead,
         float* __restrict__ out) {
    __shared__ float As[TILE_M][AS_STRIDE];
    __shared__ float zs[TILE_M][ZS_STRIDE];

    const int tid  = threadIdx.x;
    const int lane = tid & 31;
    const int wv   = tid >> 5;                // 0..3 = gate
    const int m0   = blockIdx.x * TILE_M;     // node tile base

    // ---- stage A tile = [x | Tx | h | Th] rows for 32 nodes (clamp tail) ----
    for (int i = tid; i < TILE_M * KDIM; i += 128) {
        int r = i / KDIM, cl = i % KDIM;
        int nd = m0 + r; if (nd >= NN) nd = NN - 1;   // clamp (dup rows, harmless)
        size_t node = (size_t)nd;
        float v;
        if (cl < 128)      v = x [node * IN_DIM + cl];
        else if (cl < 256) v = Tx[node * IN_DIM + (cl - 128)];
        else if (cl < 320) v = h [node * HID    + (cl - 256)];
        else               v = Th[node * HID    + (cl - 320)];
        As[r][cl] = v;
    }
    __syncthreads();

    // ---- WMMA GEMM: 32 x KDIM @ KDIM x 64 (this wave's gate columns) ----
    const int nb0 = wv * 64;
    const int row = lane & 15;            // A row (M) within a 16-row tile
    const int khalf = (lane >> 4) << 1;   // lanes 0-15 -> K+0/K+1, 16-31 -> K+2/K+3

    v8f acc[2][4];                        // [m-tile][n-tile]
    {
        #pragma unroll
        for (int t = 0; t < 4; ++t) {
            float b = bcat[nb0 + t * 16 + row];
            #pragma unroll
            for (int r8 = 0; r8 < 8; ++r8) { acc[0][t][r8] = b; acc[1][t][r8] = b; }
        }
    }

    #pragma unroll 2
    for (int k = 0; k < KDIM; k += 4) {
        const int kk = k + khalf;
        v2f a0, a1;
        a0.x = As[row][kk];       a0.y = As[row][kk + 1];
        a1.x = As[row + 16][kk];  a1.y = As[row + 16][kk + 1];
        #pragma unroll
        for (int t = 0; t < 4; ++t) {
            const int col = nb0 + t * 16 + row;
            v2f b;
            b.x = Wcat[(size_t)kk * CDIM + col];
            b.y = Wcat[(size_t)(kk + 1) * CDIM + col];
            acc[0][t] = __builtin_amdgcn_wmma_f32_16x16x4_f32(
                false, a0, false, b, (short)0, acc[0][t], false, false);
            acc[1][t] = __builtin_amdgcn_wmma_f32_16x16x4_f32(
                false, a1, false, b, (short)0, acc[1][t], false, false);
        }
    }

    // ---- spill z to LDS: C layout VGPR r, lanes<16 -> M=r, lanes>=16 -> M=r+8
    #pragma unroll
    for (int mt = 0; mt < 2; ++mt) {
        #pragma unroll
        for (int t = 0; t < 4; ++t) {
            #pragma unroll
            for (int r8 = 0; r8 < 8; ++r8) {
                int M = mt * 16 + r8 + ((lane >> 4) << 3);
                zs[M][nb0 + t * 16 + row] = acc[mt][t][r8];
            }
        }
    }
    __syncthreads();

    // ---- LSTM gating + head: 8 threads per node, 8 features each, 2 passes ----
    #pragma unroll
    for (int pass = 0; pass < 2; ++pass) {
        const int nl = pass * 16 + (tid >> 3);     // local node 0..31
        const int jb = (tid & 7) * 8;
        const int nd = m0 + nl;
        float partial = 0.0f;
        if (nd < NN) {
            const size_t node = (size_t)nd;
            #pragma unroll
            for (int jj = 0; jj < 8; ++jj) {
                int j = jb + jj;
                float cold = c[node * HID + j];
                float zi = zs[nl][0 * 64 + j];
                float zf = zs[nl][1 * 64 + j];
                float zg = zs[nl][2 * 64 + j];
                float zo = zs[nl][3 * 64 + j];
                float ig = sigm(zi + wc[0 * 64 + j] * cold);
                float fg = sigm(zf + wc[1 * 64 + j] * cold);
                float gt = tanhf(zg);
                float cn = fg * cold + ig * gt;
                float og = sigm(zo + wc[2 * 64 + j] * cn);
                float hn = og * tanhf(cn);
                out[(size_t)NN + node * HID + j] = hn;                       // h_new
                out[(size_t)NN + (size_t)NN * HID + node * HID + j] = cn;    // c_new
                float lk = (hn > 0.0f) ? hn : NEG_SLOPE * hn;
                partial += lk * Whead[j];
            }
        }
        // reduce over the 8 lanes of this node (aligned 8-lane groups in wave32)
        partial += __shfl_xor(partial, 1, 32);
        partial += __shfl_xor(partial, 2, 32);
        partial += __shfl_xor(partial, 4, 32);
        if ((tid & 7) == 0 && nd < NN) out[nd] = partial + bhead[0];
    }
}

// ------------------------------------------------------------------
// launch
// ------------------------------------------------------------------
extern "C" void kernel_launch(void* const* d_in, const int* in_sizes, int n_in,
                              void* d_out, int out_size, void* d_ws, size_t ws_size,
                              hipStream_t stream) {
    const float* x   = (const float*)d_in[0];
    const int*   ei  = (const int*)  d_in[1];
    const float* ew  = (const float*)d_in[2];
    const float* h   = (const float*)d_in[3];
    const float* c   = (const float*)d_in[4];
    const float* Wx  = (const float*)d_in[5];
    const float* bx  = (const float*)d_in[6];
    const float* Wh  = (const float*)d_in[7];
    const float* bh  = (const float*)d_in[8];
    const float* wc  = (const float*)d_in[9];
    const float* bg  = (const float*)d_in[10];
    const float* Whd = (const float*)d_in[11];
    const float* bhd = (const float*)d_in[12];

    const int* src = ei;
    const int* dst = ei + EE;

    // workspace layout (4-byte words):
    // [deg NN | cnt NN+1 | rs NN+1 | cur NN | esrc EE | enrm EE | Tx NN*128 | Th NN*64 | Wcat | bcat]
    float* ws    = (float*)d_ws;
    float* deg   = ws;                                   // NN floats
    int*   cnt   = (int*)(deg + NN);                     // NN+1 ints
    int*   rs    = cnt + (NN + 1);                       // NN+1 ints
    int*   cur   = rs + (NN + 1);                        // NN ints
    int*   esrc  = cur + NN;                             // EE ints
    float* enrm  = (float*)(esrc + EE);                  // EE floats
    float* Tx    = enrm + EE;                            // NN*128
    float* Th    = Tx + (size_t)NN * IN_DIM;             // NN*64
    float* Wcat  = Th + (size_t)NN * HID;                // KDIM*CDIM
    float* bcat  = Wcat + (size_t)KDIM * CDIM;           // CDIM

    float* out = (float*)d_out;

    // zero deg + cnt (contiguous 2*NN+1 words)
    k_zero<<<128, 256, 0, stream>>>((unsigned*)deg, (long)(2 * NN + 1));

    k_prep<<<(KDIM * CDIM + 255) / 256, 256, 0, stream>>>(Wx, Wh, bx, bh, bg, Wcat, bcat);

    k_deg_count<<<(EE + 255) / 256, 256, 0, stream>>>(src, dst, ew, deg, cnt);
    k_dinv<<<(NN + 255) / 256, 256, 0, stream>>>(deg);
    k_scan<<<1, 1024, 0, stream>>>(cnt, rs, cur);
    k_fill<<<(EE + 255) / 256, 256, 0, stream>>>(src, dst, ew, deg, cur, esrc, enrm);

    k_gather<<<NN, GCHUNK, 0, stream>>>(rs, esrc, enrm, x, h, Tx, Th);

    k_gclstm<<<(NN + TILE_M - 1) / TILE_M, 128, 0, stream>>>(
        x, Tx, h, Th, c, Wcat, bcat, wc, Whd, bhd, out);
}